// GCN_Model_90769838833737
// MI455X (gfx1250) — compile-verified
//
#include <hip/hip_runtime.h>

typedef __attribute__((ext_vector_type(2))) float v2f;
typedef __attribute__((ext_vector_type(8))) float v8f;

#define N_NODES 50000
#define N_EDGES 600000
#define D 128
#define N_LAYERS 4

// ---------------------------------------------------------------------------
// Degree counting: one thread per edge, hardware f32 atomics (L2-resident).
// ---------------------------------------------------------------------------
__global__ __launch_bounds__(256) void deg_kernel(const int* __restrict__ src,
                                                  const int* __restrict__ dst,
                                                  float* __restrict__ out_deg,
                                                  float* __restrict__ in_deg) {
    int e = blockIdx.x * blockDim.x + threadIdx.x;
    if (e < N_EDGES) {
        unsafeAtomicAdd(&out_deg[src[e]], 1.0f);
        unsafeAtomicAdd(&in_deg[dst[e]], 1.0f);
    }
}

// In-place deg -> rsqrt(max(deg,1))  (DGL clamps degree to >= 1)
__global__ __launch_bounds__(256) void norm_kernel(float* __restrict__ sn,
                                                   float* __restrict__ dn) {
    int n = blockIdx.x * blockDim.x + threadIdx.x;
    if (n < N_NODES) {
        sn[n] = rsqrtf(fmaxf(sn[n], 1.0f));
        dn[n] = rsqrtf(fmaxf(dn[n], 1.0f));
    }
}

// ---------------------------------------------------------------------------
// Edge aggregation: one wave32 per edge. Lane l handles floats [4l, 4l+4) of
// the 128-float feature row. src/dst loads are wave-uniform; the gather is a
// coalesced float4; the scatter is 4 hardware global_atomic_add_f32 per lane.
// src_norm scaling is fused into the message.
// ---------------------------------------------------------------------------
__global__ __launch_bounds__(256) void agg_kernel(const float* __restrict__ h,
                                                  const int* __restrict__ src,
                                                  const int* __restrict__ dst,
                                                  const float* __restrict__ src_norm,
                                                  float* __restrict__ agg) {
    int tid   = blockIdx.x * blockDim.x + threadIdx.x;
    int e     = tid >> 5;          // wave-uniform edge index
    int chunk = (tid & 31) * 4;    // 4 floats per lane
    if (e >= N_EDGES) return;

    int s = src[e];
    int d = dst[e];
    float sc = src_norm[s];

    const float4 v = *reinterpret_cast<const float4*>(h + (size_t)s * D + chunk);
    float* ap = agg + (size_t)d * D + chunk;
    unsafeAtomicAdd(ap + 0, v.x * sc);
    unsafeAtomicAdd(ap + 1, v.y * sc);
    unsafeAtomicAdd(ap + 2, v.z * sc);
    unsafeAtomicAdd(ap + 3, v.w * sc);
}

// ---------------------------------------------------------------------------
// out[M0+0..15, :] = (dst_norm * agg[M0+0..15, :]) @ W + bias
//
// 256 threads = 8 waves; block stages a dst_norm-scaled 16x128 A-tile into
// LDS (row pitch 132 floats -> conflict-free column-pair reads), each wave
// computes one 16(M) x 16(N) f32 tile with 32 chained
// V_WMMA_F32_16X16X4_F32 steps over K = 128.
//
// ISA 7.12.2 fragment layouts (wave32):
//   A 16x4 : lane L -> row M = L%16, K pair = (L/16)*2 .. +1   (a.x, a.y)
//   B 4x16 : lane L -> col N = L%16, K pair = (L/16)*2 .. +1   (b.x, b.y)
//   C/D    : VGPR v, lane L -> M = v + (L/16)*8, N = L%16
// ---------------------------------------------------------------------------
__global__ __launch_bounds__(256) void gemm_bias_kernel(
        const float* __restrict__ A,      // agg  [N_NODES, D]
        const float* __restrict__ dnorm,  // [N_NODES]
        const float* __restrict__ Wm,     // [D, D] (this layer)
        const float* __restrict__ bias,   // [D]   (this layer)
        float* __restrict__ out)          // [N_NODES, D]
{
    __shared__ float ldsA[16 * 132];      // 16 rows, pitch 132 (bank-conflict pad)

    const int M0 = blockIdx.x * 16;
    const int t  = threadIdx.x;

    // Cooperative stage: 2048 floats, 8 per thread, coalesced float4 loads,
    // scaled by dst_norm during staging.
#pragma unroll
    for (int half = 0; half < 2; ++half) {
        int flat = half * 1024 + t * 4;
        int row  = flat >> 7;
        int col  = flat & 127;
        float4 v = *reinterpret_cast<const float4*>(A + (size_t)(M0 + row) * D + col);
        float  s = dnorm[M0 + row];
        float* p = &ldsA[row * 132 + col];
        p[0] = v.x * s;
        p[1] = v.y * s;
        p[2] = v.z * s;
        p[3] = v.w * s;
    }
    __syncthreads();

    const int wave = t >> 5;
    const int lane = t & 31;
    const int n0   = wave * 16;            // this wave's 16 output columns
    const int mrow = lane & 15;            // A-fragment row
    const int kb   = (lane >> 4) * 2;      // A/B-fragment K sub-offset
    const int ncol = n0 + (lane & 15);     // B/D-fragment column

    v8f c = {};
#pragma unroll
    for (int k0 = 0; k0 < D; k0 += 4) {
        const float* ap = &ldsA[mrow * 132 + k0 + kb];
        v2f a;
        a.x = ap[0];
        a.y = ap[1];

        v2f b;
        b.x = Wm[(size_t)(k0 + kb    ) * D + ncol];
        b.y = Wm[(size_t)(k0 + kb + 1) * D + ncol];

        // D = A x B + C   (full f32 matrix pipe)
        c = __builtin_amdgcn_wmma_f32_16x16x4_f32(
                /*neg_a=*/false, a, /*neg_b=*/false, b,
                /*c_mod=*/(short)0, c, /*reuse_a=*/false, /*reuse_b=*/false);
    }

    const float bn    = bias[ncol];
    const int   mbase = M0 + ((lane >> 4) * 8);
#pragma unroll
    for (int v = 0; v < 8; ++v) {
        out[(size_t)(mbase + v) * D + ncol] = c[v] + bn;
    }
}

// ---------------------------------------------------------------------------
// Driver
// ---------------------------------------------------------------------------
extern "C" void kernel_launch(void* const* d_in, const int* in_sizes, int n_in,
                              void* d_out, int out_size, void* d_ws, size_t ws_size,
                              hipStream_t stream) {
    const float* feat = (const float*)d_in[0];   // [N_NODES, D]
    const int*   src  = (const int*)  d_in[1];   // [N_EDGES]
    const int*   dst  = (const int*)  d_in[2];   // [N_EDGES]
    const float* W    = (const float*)d_in[3];   // [N_LAYERS, D, D]
    const float* b    = (const float*)d_in[4];   // [N_LAYERS, D]
    float*       out  = (float*)d_out;           // [N_NODES, D]

    float* ws       = (float*)d_ws;
    float* src_norm = ws;                                   // N_NODES
    float* dst_norm = ws + N_NODES;                         // N_NODES
    float* agg      = ws + 2 * (size_t)N_NODES;             // N_NODES * D
    float* hbuf     = agg + (size_t)N_NODES * D;            // N_NODES * D

    // Degrees -> norms (deg arrays reused in place as norm arrays).
    hipMemsetAsync(src_norm, 0, 2 * (size_t)N_NODES * sizeof(float), stream);
    deg_kernel<<<(N_EDGES + 255) / 256, 256, 0, stream>>>(src, dst, src_norm, dst_norm);
    norm_kernel<<<(N_NODES + 255) / 256, 256, 0, stream>>>(src_norm, dst_norm);

    const float* h = feat;
    for (int i = 0; i < N_LAYERS; ++i) {
        hipMemsetAsync(agg, 0, (size_t)N_NODES * D * sizeof(float), stream);

        // one wave per edge: E * 32 threads
        agg_kernel<<<(N_EDGES * 32) / 256, 256, 0, stream>>>(h, src, dst, src_norm, agg);

        float* o = (i == N_LAYERS - 1) ? out : hbuf;
        gemm_bias_kernel<<<N_NODES / 16, 256, 0, stream>>>(
            agg, dst_norm, W + (size_t)i * D * D, b + (size_t)i * D, o);
        h = hbuf;
    }
}